// GCN2_83227876262524
// MI455X (gfx1250) — compile-verified
//
#include <hip/hip_runtime.h>
#include <stdint.h>

typedef __attribute__((ext_vector_type(2))) float v2f;
typedef __attribute__((ext_vector_type(8))) float v8f;

static inline int ceil_div(int a, int b) { return (a + b - 1) / b; }

// ---------------- graph normalization ----------------
__global__ void k_zero(float* __restrict__ p, int n) {
  int t = blockIdx.x * blockDim.x + threadIdx.x;
  if (t < n) p[t] = 0.0f;
}

__global__ void k_deg(const int* __restrict__ dst, float* __restrict__ deg, int E) {
  int t = blockIdx.x * blockDim.x + threadIdx.x;
  if (t < E) atomicAdd(&deg[dst[t]], 1.0f);
}

__global__ void k_dis(float* __restrict__ p, int n) {
  int t = blockIdx.x * blockDim.x + threadIdx.x;
  if (t < n) p[t] = rsqrtf(p[t] + 1.0f);  // +1 self-loop => deg >= 1
}

// ---------------- f32 WMMA GEMM (K multiple of 4, M mult of 16, NF mult of 16) ----
// V_WMMA_F32_16X16X4_F32 layout:
//   A frag: lane l<16 holds K=k0,k0+1 of row M=l; l>=16 holds K=k0+2,k0+3
//   B frag: same K split, N = lane%16
//   C/D:    vgpr r -> M = r + 8*(lane/16), N = lane%16
template <int K>
__global__ void k_gemm_wmma(const float* __restrict__ A, const float* __restrict__ W,
                            float* __restrict__ C, int M, int NF) {
  const int lane = threadIdx.x & 31;
  const int wid  = blockIdx.x * (blockDim.x >> 5) + (threadIdx.x >> 5);
  const int tilesM = M >> 4;
  const int tilesN = NF >> 4;
  if (wid >= tilesM * tilesN) return;   // wave-uniform: EXEC all-ones below
  const int mt = wid % tilesM;
  const int nt = wid / tilesM;
  const int half = lane >> 4;
  const int lm   = lane & 15;
  const int rowA = mt * 16 + lm;
  const int colB = nt * 16 + lm;
  const float* Arow = A + (size_t)rowA * K + 2 * half;   // +k0, 8B aligned pair
  const float* Wcol = W + (size_t)(2 * half) * NF + colB;

  v8f acc = {0.f, 0.f, 0.f, 0.f, 0.f, 0.f, 0.f, 0.f};
  #pragma unroll
  for (int k0 = 0; k0 < K; k0 += 4) {
    const v2f a = *reinterpret_cast<const v2f*>(Arow + k0);  // one b64 load
    v2f b;
    b.x = Wcol[(size_t)k0 * NF];
    b.y = Wcol[(size_t)k0 * NF + NF];
    acc = __builtin_amdgcn_wmma_f32_16x16x4_f32(false, a, false, b,
                                                (short)0, acc, false, false);
  }
  float* Cb = C + (size_t)(mt * 16 + 8 * half) * NF + colB;
  #pragma unroll
  for (int r = 0; r < 8; ++r) Cb[(size_t)r * NF] = acc[r];
}

// K == 3 special case (layer 1): single WMMA, branchless zero-pad to K=4.
__global__ void k_gemm3_wmma(const float* __restrict__ A, const float* __restrict__ W,
                             float* __restrict__ C, int M, int NF) {
  const int lane = threadIdx.x & 31;
  const int wid  = blockIdx.x * (blockDim.x >> 5) + (threadIdx.x >> 5);
  const int tilesM = M >> 4;
  const int tilesN = NF >> 4;
  if (wid >= tilesM * tilesN) return;
  const int mt = wid % tilesM;
  const int nt = wid / tilesM;
  const int half = lane >> 4;      // 0: holds K=0,1   1: holds K=2,(3->0)
  const int lm   = lane & 15;
  const int rowA = mt * 16 + lm;
  const int colB = nt * 16 + lm;

  v2f a, b;
  a.x = A[rowA * 3 + 2 * half];          // K=0 or K=2 (always in-bounds)
  const float a1 = A[rowA * 3 + 1];      // K=1 (always in-bounds)
  a.y = half ? 0.0f : a1;                // K=3 padded with 0 via cndmask
  b.x = W[(size_t)(2 * half) * NF + colB];
  const float b1 = W[(size_t)NF + colB];
  b.y = half ? 0.0f : b1;

  v8f acc = {0.f, 0.f, 0.f, 0.f, 0.f, 0.f, 0.f, 0.f};
  acc = __builtin_amdgcn_wmma_f32_16x16x4_f32(false, a, false, b,
                                              (short)0, acc, false, false);
  float* Cb = C + (size_t)(mt * 16 + 8 * half) * NF + colB;
  #pragma unroll
  for (int r = 0; r < 8; ++r) Cb[(size_t)r * NF] = acc[r];
}

// ---------------- aggregation ----------------
// agg[i,f] = dis[i]^2 * xw[i,f] + bias[f]   (self-loop + bias; fully writes agg)
__global__ void k_init_agg(const float* __restrict__ xw, const float* __restrict__ dis,
                           const float* __restrict__ bias, float* __restrict__ agg,
                           int total, int lf) {
  int t = blockIdx.x * blockDim.x + threadIdx.x;
  if (t >= total) return;
  const int i = t >> lf;
  const int f = t & ((1 << lf) - 1);
  const float d = dis[i];
  agg[t] = d * d * xw[t] + bias[f];
}

// edge scatter: one thread per (edge, 4-feature group)
__global__ void k_scatter(const int* __restrict__ src, const int* __restrict__ dst,
                          const float* __restrict__ dis, const float* __restrict__ xw,
                          float* __restrict__ agg, int E, int F, int lf4) {
  const int t = blockIdx.x * blockDim.x + threadIdx.x;
  const int total = E << lf4;
  if (t >= total) return;
  const int e = t >> lf4;
  const int q = (t & ((1 << lf4) - 1)) << 2;
  const int s = src[e];
  const int d = dst[e];
  const float nrm = dis[s] * dis[d];
  const float4 v = *reinterpret_cast<const float4*>(xw + s * F + q);
  float* base = agg + d * F + q;
  atomicAdd(base + 0, nrm * v.x);
  atomicAdd(base + 1, nrm * v.y);
  atomicAdd(base + 2, nrm * v.z);
  atomicAdd(base + 3, nrm * v.w);
}

// ---------------- batchnorm ----------------
// grid = F blocks of 256: 256 accumulator slots per channel
__global__ void k_bn_stats(const float* __restrict__ agg, float* __restrict__ stats,
                           int N, int F, int lf) {
  const int t = blockIdx.x * blockDim.x + threadIdx.x;
  const int f = t & (F - 1);
  const int slot = t >> lf;
  float s = 0.f, s2 = 0.f;
  for (int r = slot; r < N; r += 256) {
    const float v = agg[r * F + f];
    s += v; s2 += v * v;
  }
  atomicAdd(&stats[f], s);
  atomicAdd(&stats[F + f], s2);
}

__global__ void k_bn_fin(const float* __restrict__ stats, const float* __restrict__ g,
                         const float* __restrict__ be, float* __restrict__ scsh,
                         int F, float invN) {
  const int f = blockIdx.x * blockDim.x + threadIdx.x;
  if (f >= F) return;
  const float mean = stats[f] * invN;
  const float var  = stats[F + f] * invN - mean * mean;
  const float sc = g[f] * rsqrtf(var + 1e-5f);
  scsh[f] = sc;
  scsh[F + f] = be[f] - mean * sc;
}

// in-place BN apply + leaky relu
__global__ void k_bn_lrelu(float* __restrict__ agg, const float* __restrict__ scsh,
                           int total, int F) {
  const int t = blockIdx.x * blockDim.x + threadIdx.x;
  if (t >= total) return;
  const int f = t & (F - 1);
  const float v = scsh[f] * agg[t] + scsh[F + f];
  agg[t] = v > 0.f ? v : 0.1f * v;
}

// ---------------- layer 4 (out dim == 1) ----------------
__global__ void k_gemv256(const float* __restrict__ h, const float* __restrict__ w,
                          float* __restrict__ z, int N) {
  const int wave = blockIdx.x * (blockDim.x >> 5) + (threadIdx.x >> 5);
  const int lane = threadIdx.x & 31;
  if (wave >= N) return;
  const float* row = h + wave * 256;
  float s = 0.f;
  #pragma unroll
  for (int j = 0; j < 8; ++j) s += row[lane + 32 * j] * w[lane + 32 * j];
  #pragma unroll
  for (int off = 16; off >= 1; off >>= 1) s += __shfl_xor(s, off, 32);
  if (lane == 0) z[wave] = s;
}

__global__ void k_init1(const float* __restrict__ z, const float* __restrict__ dis,
                        const float* __restrict__ b, float* __restrict__ agg, int N) {
  const int t = blockIdx.x * blockDim.x + threadIdx.x;
  if (t < N) { const float d = dis[t]; agg[t] = d * d * z[t] + b[0]; }
}

__global__ void k_scatter1(const int* __restrict__ src, const int* __restrict__ dst,
                           const float* __restrict__ dis, const float* __restrict__ z,
                           float* __restrict__ agg, int E) {
  const int t = blockIdx.x * blockDim.x + threadIdx.x;
  if (t >= E) return;
  const int s = src[t], d = dst[t];
  atomicAdd(&agg[d], dis[s] * dis[d] * z[s]);
}

__global__ void k_sigmoid(const float* __restrict__ a, float* __restrict__ out, int N) {
  const int t = blockIdx.x * blockDim.x + threadIdx.x;
  if (t < N) out[t] = 1.0f / (1.0f + expf(-a[t]));
}

// ---------------- one GCN+BN+LReLU layer (post-GEMM part) ----------------
static void gcn_post(const float* bias, const float* g, const float* be, int F, int lf,
                     const int* src, const int* dstp, const float* dis,
                     float* X, float* G, float* stats, float* scsh,
                     int N, int E, hipStream_t stream) {
  const int total = N * F;
  k_init_agg<<<ceil_div(total, 256), 256, 0, stream>>>(X, dis, bias, G, total, lf);
  const int stotal = E << (lf - 2);
  k_scatter<<<ceil_div(stotal, 256), 256, 0, stream>>>(src, dstp, dis, X, G, E, F, lf - 2);
  k_zero<<<ceil_div(2 * F, 256), 256, 0, stream>>>(stats, 2 * F);
  k_bn_stats<<<F, 256, 0, stream>>>(G, stats, N, F, lf);
  k_bn_fin<<<1, 256, 0, stream>>>(stats, g, be, scsh, F, 1.0f / (float)N);
  k_bn_lrelu<<<ceil_div(total, 256), 256, 0, stream>>>(G, scsh, total, F);
}

extern "C" void kernel_launch(void* const* d_in, const int* in_sizes, int n_in,
                              void* d_out, int out_size, void* d_ws, size_t ws_size,
                              hipStream_t stream) {
  const float* x   = (const float*)d_in[0];
  const int*   ei  = (const int*)  d_in[1];
  const float* W1  = (const float*)d_in[2];
  const float* b1  = (const float*)d_in[3];
  const float* g1  = (const float*)d_in[4];
  const float* be1 = (const float*)d_in[5];
  const float* W2  = (const float*)d_in[6];
  const float* b2  = (const float*)d_in[7];
  const float* g2  = (const float*)d_in[8];
  const float* be2 = (const float*)d_in[9];
  const float* W3  = (const float*)d_in[10];
  const float* b3  = (const float*)d_in[11];
  const float* g3  = (const float*)d_in[12];
  const float* be3 = (const float*)d_in[13];
  const float* W4  = (const float*)d_in[14];
  const float* b4  = (const float*)d_in[15];

  const int N = in_sizes[0] / 3;
  const int E = in_sizes[1] / 2;
  const int* src  = ei;
  const int* dstp = ei + E;

  float* ws    = (float*)d_ws;
  float* dis   = ws;                                       // N floats
  float* stats = ws + (((size_t)N + 255) & ~(size_t)255);  // 512 floats
  float* scsh  = stats + 512;                              // 512 floats
  float* X     = (float*)((((uintptr_t)(scsh + 512)) + 1023) & ~(uintptr_t)1023);
  const size_t bufE = (size_t)N * 256;
  float* G     = X + bufE;                                 // total ~103 MB of d_ws

  // graph normalization (shared by all layers)
  k_zero<<<ceil_div(N, 256), 256, 0, stream>>>(dis, N);
  k_deg <<<ceil_div(E, 256), 256, 0, stream>>>(dstp, dis, E);
  k_dis <<<ceil_div(N, 256), 256, 0, stream>>>(dis, N);

  // layer 1: x[N,3] @ W1[3,64]
  {
    const int tiles = (N >> 4) * (64 >> 4);
    k_gemm3_wmma<<<ceil_div(tiles, 8), 256, 0, stream>>>(x, W1, X, N, 64);
    gcn_post(b1, g1, be1, 64, 6, src, dstp, dis, X, G, stats, scsh, N, E, stream);
  }
  // layer 2: h[N,64] @ W2[64,64]
  {
    const int tiles = (N >> 4) * (64 >> 4);
    k_gemm_wmma<64><<<ceil_div(tiles, 8), 256, 0, stream>>>(G, W2, X, N, 64);
    gcn_post(b2, g2, be2, 64, 6, src, dstp, dis, X, G, stats, scsh, N, E, stream);
  }
  // layer 3: h[N,64] @ W3[64,256]
  {
    const int tiles = (N >> 4) * (256 >> 4);
    k_gemm_wmma<64><<<ceil_div(tiles, 8), 256, 0, stream>>>(G, W3, X, N, 256);
    gcn_post(b3, g3, be3, 256, 8, src, dstp, dis, X, G, stats, scsh, N, E, stream);
  }
  // layer 4: GEMV(256->1) + GCN aggregation + sigmoid
  float* z = X;  // xw buffer is free again
  k_gemv256 <<<ceil_div(N, 8),   256, 0, stream>>>(G, W4, z, N);
  k_init1   <<<ceil_div(N, 256), 256, 0, stream>>>(z, dis, b4, G, N);
  k_scatter1<<<ceil_div(E, 256), 256, 0, stream>>>(src, dstp, dis, z, G, E);
  k_sigmoid <<<ceil_div(N, 256), 256, 0, stream>>>(G, (float*)d_out, N);
}